// RNNCoAttention_62139586839198
// MI455X (gfx1250) — compile-verified
//
#include <hip/hip_runtime.h>
#include <math.h>

typedef __attribute__((ext_vector_type(2))) float v2f;
typedef __attribute__((ext_vector_type(8))) float v8f;

#define B_   32
#define T_   1024
#define D_   256
#define H_   256
#define G4_  1024          // 4*H
#define H2_  512           // 2*H
#define ROW_ 1280          // 2H (h) + 2H (attn) + D (x)
#define BT_  (B_*T_)       // 32768
#define EPSC 1e-8f
#define WPITCH 260         // LDS row pitch for Whh slice: +4 word skew -> no bank conflicts

// ---- f32 WMMA 16x16x4 (f32 in, f32 accumulate) -----------------------------
// A 16x4: lanes 0-15 hold M=lane, K={0,1}; lanes 16-31 hold M=lane-16, K={2,3}
// B 4x16: lanes 0-15 hold N=lane, K={0,1}; lanes 16-31 hold N=lane-16, K={2,3}
// C/D 16x16: VGPR r: lanes 0-15 -> (M=r, N=lane); lanes 16-31 -> (M=8+r, N=lane-16)
__device__ __forceinline__ v8f wmma4(v2f a, v2f b, v8f c) {
  return __builtin_amdgcn_wmma_f32_16x16x4_f32(false, a, false, b, (short)0, c,
                                               false, false);
}

__device__ __forceinline__ float sigm(float x) { return 1.0f / (1.0f + __expf(-x)); }

// low 32 bits of a flat shared-aperture address == LDS byte offset
__device__ __forceinline__ unsigned lds_u32(const void* p) {
  return (unsigned)(unsigned long long)p;
}

// ASYNCcnt-tracked global -> LDS copy, 16 bytes per lane (no VGPR bounce)
__device__ __forceinline__ void async_load_b128(unsigned lds_off, const void* gaddr) {
  asm volatile("global_load_async_to_lds_b128 %0, %1, off"
               :: "v"(lds_off), "v"(gaddr)
               : "memory");
}
__device__ __forceinline__ void wait_asynccnt0() {
  asm volatile("s_wait_asynccnt 0x0" ::: "memory");
}

// ============================================================================
// Kernel 1: x_proj = x @ Wih^T + b  for z = {xp,xq} x {fwd,bwd}
// One wave per 16x16 tile of the (32768 x 1024) output, K = 256.
// ============================================================================
__global__ __launch_bounds__(256)
void xproj_kernel(const float* __restrict__ xp, const float* __restrict__ xq,
                  const float* __restrict__ Wih_f, const float* __restrict__ b_f,
                  const float* __restrict__ Wih_b, const float* __restrict__ b_b,
                  float* __restrict__ ws_xproj) {
  const int z = blockIdx.z;                        // 0:xp-f 1:xp-b 2:xq-f 3:xq-b
  const float* x   = (z < 2) ? xp : xq;
  const float* W   = (z & 1) ? Wih_b : Wih_f;      // (4H, D) row-major
  const float* bia = (z & 1) ? b_b   : b_f;
  float* out = ws_xproj + (size_t)z * BT_ * G4_;

  const int lane  = threadIdx.x;
  const int lin   = blockIdx.x * 8 + threadIdx.y;  // 131072 tiles total
  const int nt    = lin & 63;                      // 1024/16 N tiles
  const int mt    = lin >> 6;                      // 32768/16 M tiles
  const int m0 = mt * 16, n0 = nt * 16;

  const int l15   = lane & 15;
  const int khalf = (lane >> 4) * 2;

  v8f acc = {0.f, 0.f, 0.f, 0.f, 0.f, 0.f, 0.f, 0.f};
  const float* arow = x + (size_t)(m0 + l15) * D_ + khalf;  // A[m][k]
  const float* brow = W + (size_t)(n0 + l15) * D_ + khalf;  // B[k][n] = Wih[n][k]
  for (int kb = 0; kb < D_; kb += 4) {
    v2f a = *(const v2f*)(arow + kb);
    v2f b = *(const v2f*)(brow + kb);
    acc = wmma4(a, b, acc);
  }

  const int c     = n0 + l15;
  const int rbase = m0 + ((lane >> 4) * 8);
  const float bv  = bia[c];
#pragma unroll
  for (int r = 0; r < 8; ++r)
    out[(size_t)(rbase + r) * G4_ + c] = acc[r] + bv;
}

// ============================================================================
// Kernel 2: 4 LSTM recurrences (xp/xq x fwd/bwd), 32 WGs per recurrence.
// Each WG owns 8 hidden units -> 32 gate columns.  The Whh slice (32 rows x
// 256) is staged into LDS via ASYNC global->LDS B128 copies (row-major with
// a 260-float pitch so B-fragment reads are bank-conflict free ds_load_b64).
// h (32x256) lives in a double-buffered global region; one inter-WG barrier
// per timestep (atomic arrive counter + agent fences).  Next timestep's
// x_proj gate slices are prefetched (global_prefetch_b8) to overlap HBM
// latency with the barrier spin.  Writes h_t into output cols [dir*H, +H).
// ============================================================================
__global__ __launch_bounds__(128)
void lstm_recur_kernel(const float* __restrict__ ws_xproj,
                       const float* __restrict__ Whh_f,
                       const float* __restrict__ Whh_b,
                       float* __restrict__ out,
                       float* __restrict__ hbufs,
                       unsigned* __restrict__ cnts) {
  __shared__ float sW[32 * WPITCH];  // sW[c*WPITCH + k] = Whh[row(c)][k]
  __shared__ float sG[32 * 32];      // gates: [batch][g*8+jj]
  __shared__ float sC[32 * 8];       // cell state slice

  const int rec = blockIdx.y;        // 0:xp-f 1:xp-b 2:xq-f 3:xq-b
  const int wg  = blockIdx.x;        // 0..31
  const int dir = rec & 1;
  const int inp = rec >> 1;
  const int j0  = wg * 8;

  const float* xproj = ws_xproj + (size_t)rec * BT_ * G4_;
  const float* Whh   = dir ? Whh_b : Whh_f;          // (4H, H) row-major
  float* outp = out + (size_t)inp * BT_ * ROW_;
  float* hb0  = hbufs + (size_t)rec * 2 * (B_ * H_);
  float* hb1  = hb0 + B_ * H_;
  unsigned* cnt = cnts + rec * 32;                   // padded counters

  const int lane = threadIdx.x;
  const int wv   = threadIdx.y;                      // 0..3
  const int tid  = wv * 32 + lane;

  // ---- async global -> LDS stage of the Whh slice (2048 x B128 transfers) ----
  // column c (0..31) -> Whh row (c/8)*256 + j0 + (c%8); 64 16-byte segs per row
  for (int i = tid; i < 32 * 64; i += 128) {
    int c = i >> 6, seg = i & 63;
    int row = ((c >> 3) * 256) + j0 + (c & 7);
    const float* g = Whh + (size_t)row * H_ + seg * 4;
    async_load_b128(lds_u32(&sW[c * WPITCH + seg * 4]), g);
  }
  for (int i = tid; i < 256; i += 128) sC[i] = 0.f;
  wait_asynccnt0();
  __syncthreads();

  const int mt = wv & 1, ntw = wv >> 1;
  const int m0 = mt * 16, n0 = ntw * 16;
  const int l15   = lane & 15;
  const int khalf = (lane >> 4) * 2;
  const int ccol  = n0 + l15;
  const float* sWrow = sW + ccol * WPITCH + khalf;

  for (int s = 0; s < T_; ++s) {
    const int t = dir ? (T_ - 1 - s) : s;
    const float* hprev = (s & 1) ? hb1 : hb0;
    float* hnext       = (s & 1) ? hb0 : hb1;

    // ---- gates_tile = h_prev @ Whh_slice^T  (M=32, N=32, K=256) ----
    v8f acc = {0.f, 0.f, 0.f, 0.f, 0.f, 0.f, 0.f, 0.f};
    const float* arow = hprev + (m0 + l15) * H_ + khalf;
    for (int kb = 0; kb < H_; kb += 4) {
      v2f a = *(const v2f*)(arow + kb);
      v2f b = *(const v2f*)(sWrow + kb);   // ds_load_b64, bank-conflict free
      acc = wmma4(a, b, acc);
    }
    {
      const int rb = m0 + ((lane >> 4) * 8);
#pragma unroll
      for (int r = 0; r < 8; ++r) sG[(rb + r) * 32 + ccol] = acc[r];
    }
    __syncthreads();

    // ---- elementwise gate math; update c, emit h ----
    for (int e = tid; e < 256; e += 128) {
      int bb = e >> 3, jj = e & 7;
      size_t base = ((size_t)(bb * T_ + t)) * G4_ + j0 + jj;
      float gi = sG[bb * 32 +      jj] + xproj[base];
      float gf = sG[bb * 32 +  8 + jj] + xproj[base + 256];
      float gg = sG[bb * 32 + 16 + jj] + xproj[base + 512];
      float go = sG[bb * 32 + 24 + jj] + xproj[base + 768];
      float cn = sigm(gf) * sC[e] + sigm(gi) * tanhf(gg);
      float hv = sigm(go) * tanhf(cn);
      sC[e] = cn;
      hnext[bb * H_ + j0 + jj] = hv;
      outp[((size_t)(bb * T_ + t)) * ROW_ + dir * H_ + j0 + jj] = hv;

      // prefetch next step's gate inputs (overlaps the barrier spin)
      if (s + 1 < T_) {
        int tn = dir ? (t - 1) : (t + 1);
        size_t basen = ((size_t)(bb * T_ + tn)) * G4_ + j0 + jj;
        __builtin_prefetch(xproj + basen,       0, 1);
        __builtin_prefetch(xproj + basen + 256, 0, 1);
        __builtin_prefetch(xproj + basen + 512, 0, 1);
        __builtin_prefetch(xproj + basen + 768, 0, 1);
      }
    }
    __builtin_amdgcn_fence(__ATOMIC_RELEASE, "agent");   // publish h slice
    __syncthreads();

    // ---- inter-workgroup step barrier (32 WGs of this recurrence) ----
    if (tid == 0) {
      __hip_atomic_fetch_add(cnt, 1u, __ATOMIC_RELAXED, __HIP_MEMORY_SCOPE_AGENT);
      const unsigned target = 32u * (unsigned)(s + 1);
      while (__hip_atomic_load(cnt, __ATOMIC_RELAXED, __HIP_MEMORY_SCOPE_AGENT) < target)
        __builtin_amdgcn_s_sleep(2);
    }
    __syncthreads();
    __builtin_amdgcn_fence(__ATOMIC_ACQUIRE, "agent");   // see peers' h slices
  }
}

// ============================================================================
// Kernel 3: row L2-norms of hp / hq (over the 2H h-columns of the output).
// ============================================================================
__global__ __launch_bounds__(256)
void norm_kernel(const float* __restrict__ out, float* __restrict__ nrm) {
  int row  = blockIdx.x * 8 + threadIdx.y;   // 0..65535 (p rows then q rows)
  int lane = threadIdx.x;
  const int side = row >= BT_;
  const int r    = side ? row - BT_ : row;
  const float* h = out + (size_t)side * BT_ * ROW_ + (size_t)r * ROW_;
  float s = 0.f;
  for (int k = lane; k < H2_; k += 32) { float v = h[k]; s += v * v; }
#pragma unroll
  for (int off = 16; off > 0; off >>= 1) s += __shfl_xor(s, off, 32);
  if (lane == 0) nrm[row] = sqrtf(s);
}

// ============================================================================
// Kernel 4: flash-style cosine co-attention.  One WG per (side, batch, 16-row
// tile).  Scores for a 512-key chunk live in 32 KB LDS; online softmax across
// the two chunks; weighted-sum accumulators stay in registers (v8f x4 / wave).
// ============================================================================
__global__ __launch_bounds__(256)
void attn_kernel(float* __restrict__ out, const float* __restrict__ nrm) {
  __shared__ float sS[16 * 512];
  __shared__ float sM[16], sL[16], sScale[16];

  const int b    = blockIdx.y;
  const int i0   = blockIdx.x * 16;
  const int side = blockIdx.z;               // 0: p rows / q keys, 1: swapped

  float* Qbase       = out + (size_t)side * BT_ * ROW_;
  const float* Kbase = out + (size_t)(1 - side) * BT_ * ROW_;
  const float* nQ = nrm + (size_t)side * BT_;
  const float* nK = nrm + (size_t)(1 - side) * BT_;

  const int lane = threadIdx.x, wv = threadIdx.y;
  const int tid  = wv * 32 + lane;
  const int l15   = lane & 15;
  const int khalf = (lane >> 4) * 2;

  if (tid < 16) { sM[tid] = -INFINITY; sL[tid] = 0.f; }
  __syncthreads();

  v8f oacc[4];
#pragma unroll
  for (int q = 0; q < 4; ++q) {
    v8f z = {0.f, 0.f, 0.f, 0.f, 0.f, 0.f, 0.f, 0.f};
    oacc[q] = z;
  }

  const float* Qrow = Qbase + ((size_t)(b * T_ + i0 + l15)) * ROW_;  // h cols 0..511

  for (int ch = 0; ch < 2; ++ch) {
    const int jc = ch * 512;

    // ---- phase 1: cosine scores -> sS (16 x 512) ----
#pragma unroll 1
    for (int q = 0; q < 4; ++q) {
      const int ntile = wv + q * 8;                 // 0..31
      const int jcol  = ntile * 16 + l15;
      const float* Krow = Kbase + ((size_t)(b * T_ + jc + jcol)) * ROW_;
      v8f acc = {0.f, 0.f, 0.f, 0.f, 0.f, 0.f, 0.f, 0.f};
      for (int kb = 0; kb < H2_; kb += 4) {
        v2f a  = *(const v2f*)(Qrow + khalf + kb);
        v2f bb = *(const v2f*)(Krow + khalf + kb);
        acc = wmma4(a, bb, acc);
      }
      const float nkv = nK[b * T_ + jc + jcol];
      const int rb = (lane >> 4) * 8;
#pragma unroll
      for (int r = 0; r < 8; ++r) {
        float nqv = nQ[b * T_ + i0 + rb + r];
        sS[(rb + r) * 512 + ntile * 16 + l15] = acc[r] / fmaxf(nqv * nkv, EPSC);
      }
    }
    __syncthreads();

    // ---- online softmax bookkeeping ----
    if (tid < 16) {
      float mc = -INFINITY;
      const float* rowp = sS + tid * 512;
      for (int k2 = 0; k2 < 512; ++k2) mc = fmaxf(mc, rowp[k2]);
      float mold = sM[tid];
      float mnew = fmaxf(mold, mc);
      sScale[tid] = (mold == -INFINITY) ? 0.f : __expf(mold - mnew);
      sM[tid] = mnew;
    }
    __syncthreads();
    for (int idx = tid; idx < 16 * 512; idx += 256) {
      int rrow = idx >> 9;
      sS[idx] = __expf(sS[idx] - sM[rrow]);
    }
    __syncthreads();
    if (tid < 16) {
      float sum = 0.f;
      const float* rowp = sS + tid * 512;
      for (int k2 = 0; k2 < 512; ++k2) sum += rowp[k2];
      sL[tid] = sL[tid] * sScale[tid] + sum;
    }
    {
      const int rb = (lane >> 4) * 8;
#pragma unroll
      for (int q = 0; q < 4; ++q)
#pragma unroll
        for (int r = 0; r < 8; ++r) oacc[q][r] *= sScale[rb + r];
    }

    // ---- phase 3: oacc += P @ V  (K = 512 keys of this chunk) ----
#pragma unroll 1
    for (int q = 0; q < 4; ++q) {
      const int ncol = (wv + q * 8) * 16 + l15;     // feature column 0..511
      v8f acc = oacc[q];
      const float* Prow = sS + l15 * 512 + khalf;
      for (int kb = 0; kb < 512; kb += 4) {
        v2f a = *(const v2f*)(Prow + kb);
        int kk = jc + kb + khalf;
        v2f bb;
        bb.x = Kbase[((size_t)(b * T_ + kk)) * ROW_ + ncol];
        bb.y = Kbase[((size_t)(b * T_ + kk + 1)) * ROW_ + ncol];
        acc = wmma4(a, bb, acc);
      }
      oacc[q] = acc;
    }
    __syncthreads();
  }

  // ---- final: normalize and write attn block (cols [2H, 4H)) ----
  const int rb2 = (lane >> 4) * 8;
#pragma unroll
  for (int q = 0; q < 4; ++q) {
    const int ncol = (wv + q * 8) * 16 + l15;
#pragma unroll
    for (int r = 0; r < 8; ++r)
      Qbase[((size_t)(b * T_ + i0 + rb2 + r)) * ROW_ + H2_ + ncol] =
          oacc[q][r] / sL[rb2 + r];
  }
}

// ============================================================================
// Kernel 5: copy raw inputs into output columns [4H, 4H+D)
// ============================================================================
__global__ void copyx_kernel(const float* __restrict__ xp,
                             const float* __restrict__ xq,
                             float* __restrict__ out) {
  size_t idx = (size_t)blockIdx.x * blockDim.x + threadIdx.x;
  const size_t half = (size_t)BT_ * D_;
  if (idx >= 2 * half) return;
  int side = idx >= half;
  size_t rem = idx - (size_t)side * half;
  size_t row = rem / D_;
  int col = (int)(rem % D_);
  const float* src = side ? xq : xp;
  out[(size_t)side * BT_ * ROW_ + row * ROW_ + 1024 + col] = src[rem];
}

// ============================================================================
extern "C" void kernel_launch(void* const* d_in, const int* in_sizes, int n_in,
                              void* d_out, int out_size, void* d_ws, size_t ws_size,
                              hipStream_t stream) {
  const float* xp    = (const float*)d_in[0];
  const float* xq    = (const float*)d_in[1];
  const float* Wih_f = (const float*)d_in[2];
  const float* Whh_f = (const float*)d_in[3];
  const float* b_f   = (const float*)d_in[4];
  const float* Wih_b = (const float*)d_in[5];
  const float* Whh_b = (const float*)d_in[6];
  const float* b_b   = (const float*)d_in[7];
  float* out = (float*)d_out;
  float* ws  = (float*)d_ws;

  // workspace layout (float offsets)
  const size_t XPROJ = 0;                                  // 4 * 32768*1024
  const size_t HB    = (size_t)4 * BT_ * G4_;              // 4 recs * 2 * 32*256
  const size_t NRM   = HB + (size_t)4 * 2 * (B_ * H_);     // 2 * BT norms
  const size_t CNT   = NRM + (size_t)2 * BT_;              // 4 * 32 uint counters
  const size_t ZEND  = CNT + 4 * 32;

  // zero h double-buffers + barrier counters (capture-safe memset node)
  hipMemsetAsync((char*)d_ws + HB * sizeof(float), 0,
                 (ZEND - HB) * sizeof(float), stream);

  xproj_kernel<<<dim3(16384, 1, 4), dim3(32, 8), 0, stream>>>(
      xp, xq, Wih_f, b_f, Wih_b, b_b, ws + XPROJ);

  lstm_recur_kernel<<<dim3(32, 4), dim3(32, 4), 0, stream>>>(
      ws + XPROJ, Whh_f, Whh_b, out, ws + HB, (unsigned*)(ws + CNT));

  norm_kernel<<<dim3(8192), dim3(32, 8), 0, stream>>>(out, ws + NRM);

  attn_kernel<<<dim3(64, 32, 2), dim3(32, 8), 0, stream>>>(out, ws + NRM);

  copyx_kernel<<<dim3((2 * BT_ * D_ + 255) / 256), dim3(256), 0, stream>>>(
      xp, xq, out);
}